// TopKJSDivergence_58059367907782
// MI455X (gfx1250) — compile-verified
//
#include <hip/hip_runtime.h>
#include <stdint.h>
#include <utility>

#define VV     50257
#define KTOP   50
#define EPSF   1e-8f
#define BLOCK  512
#define CAP    2048            // stage-1 candidate capacity
#define CAP2   256             // stage-2 candidate capacity
#define N4MAX  (VV >> 2)                          // 12564 float4 blocks (max)
#define NITER  ((N4MAX + BLOCK - 1) / BLOCK)      // 25 b128 issues per wave

// ---------------- CDNA5 async global->LDS copy (ASYNCcnt path) ----------------
__device__ __forceinline__ void async_copy_b32(uint32_t lds_byte, const float* g) {
    asm volatile("global_load_async_to_lds_b32 %0, %1, off"
                 :: "v"(lds_byte), "v"((uint64_t)(uintptr_t)g) : "memory");
}
__device__ __forceinline__ void async_copy_b128(uint32_t lds_byte, const float* g) {
    asm volatile("global_load_async_to_lds_b128 %0, %1, off"
                 :: "v"(lds_byte), "v"((uint64_t)(uintptr_t)g) : "memory");
}
template <int N>
__device__ __forceinline__ void asm_wait_async() {            // constant immediate
    asm volatile("s_wait_asynccnt %0" :: "n"(N) : "memory");
}

// Issue the whole row as async copies. EVERY wave issues exactly 27 async
// instructions (head b32, tail b32, NITER clamped b128) so per-wave ASYNCcnt
// accounting is uniform; redundant clamped issues rewrite the same data.
__device__ __forceinline__ void issue_row_loads(const float* __restrict__ grow,
                                                uint32_t ldsBase, int s, int h,
                                                int n4, int ntail, int t0) {
    const int tid = threadIdx.x;
    {   // head (elements [0,h)), redundant lanes reload element 0
        const int e = (tid < h) ? tid : 0;
        async_copy_b32(ldsBase + (uint32_t)((e + s) << 2), grow + e);
    }
    {   // tail (elements [t0,VV)), redundant lanes reload element 0
        const int e = (tid < ntail) ? (t0 + tid) : 0;
        async_copy_b32(ldsBase + (uint32_t)((e + s) << 2), grow + e);
    }
    for (int k = 0; k < NITER; ++k) {             // b128 bulk, index-clamped
        int i = tid + k * BLOCK;
        if (i > n4 - 1) i = n4 - 1;
        const int e = h + (i << 2);
        async_copy_b128(ldsBase + (uint32_t)((e + s) << 2), grow + e);
    }
}

// ---------------- L1 histogram pipelined against the async stream ----------------
__device__ __forceinline__ void hist_chunk(int k, int tid, const float* data,
                                           const uint4* rb4, int h, int n4,
                                           int ntail, int t0, uint32_t* hist13) {
    if (k == 0) {   // head/tail were each wave's first issues -> already complete
        if (tid < h)     atomicAdd(&hist13[__float_as_uint(data[tid]) >> 19], 1u);
        if (tid < ntail) atomicAdd(&hist13[__float_as_uint(data[t0 + tid]) >> 19], 1u);
    }
    const int i = tid + k * BLOCK;
    if (i < n4) {
        const uint4 v = rb4[i];
        atomicAdd(&hist13[v.x >> 19], 1u);
        atomicAdd(&hist13[v.y >> 19], 1u);
        atomicAdd(&hist13[v.z >> 19], 1u);
        atomicAdd(&hist13[v.w >> 19], 1u);
    }
}
template <int K2>
__device__ __forceinline__ void hist_pipe(int tid, const float* data, const uint4* rb4,
                                          int h, int n4, int ntail, int t0,
                                          uint32_t* hist13) {
    if constexpr (K2 < NITER) {
        // chunk K2 of this wave done when <= (NITER-1-K2) of its 27 issues remain
        asm_wait_async<NITER - 1 - K2>();
        hist_chunk(K2, tid, data, rb4, h, n4, ntail, t0, hist13);
        hist_pipe<K2 + 1>(tid, data, rb4, h, n4, ntail, t0, hist13);
    }
}

// -------- 3-level scan of the 8192-bin histogram -> threshold bin (<=64 serial) --------
__device__ __forceinline__ void scan13(uint32_t* hist13, uint32_t* grpA,  // 512
                                       uint32_t* grpB,                     // 32
                                       uint32_t* shv) {
    const int tid = threadIdx.x;
    {   // level A: 512 partial sums of 16 bins each
        uint32_t ssum = 0;
        const uint32_t* hb = hist13 + (tid << 4);
        for (int j = 0; j < 16; ++j) ssum += hb[j];
        grpA[tid] = ssum;
    }
    __syncthreads();
    if (tid < 32) {   // level B: 32 sums of 16 grpA each
        uint32_t ssum = 0;
        const uint32_t* gb = grpA + (tid << 4);
        for (int j = 0; j < 16; ++j) ssum += gb[j];
        grpB[tid] = ssum;
    }
    __syncthreads();
    if (tid == 0) {
        uint32_t acc = 0; int a = 31;
        for (; a > 0; --a) { const uint32_t c = grpB[a]; if (acc + c >= (uint32_t)KTOP) break; acc += c; }
        int g = (a << 4) + 15;
        for (; g > (a << 4); --g) { const uint32_t c = grpA[g]; if (acc + c >= (uint32_t)KTOP) break; acc += c; }
        int b = (g << 4) + 15;
        for (; b > (g << 4); --b) { const uint32_t c = hist13[b]; if (acc + c >= (uint32_t)KTOP) break; acc += c; }
        shv[0] = (uint32_t)b;                // b0
        shv[1] = acc;                        // strictly-above count
        shv[2] = ((uint32_t)b) << 19;        // threshold T1
        shv[3] = 0;                          // candidate counter
    }
    __syncthreads();
}

// ---------------- candidate collection (wave-aggregated pushes) ----------------
__device__ __forceinline__ void collect_chunk(int k, int tid, const uint4* rb4, int h,
                                              int n4, uint32_t T1, uint32_t* shcnt,
                                              uint32_t* candKey, uint32_t* candIdx) {
    const int i = tid + k * BLOCK;
    const bool in = (i < n4);
    uint4 v = make_uint4(0u, 0u, 0u, 0u);
    if (in) v = rb4[i];
    const uint32_t m0 = (in && v.x >= T1) ? 1u : 0u;
    const uint32_t m1 = (in && v.y >= T1) ? 1u : 0u;
    const uint32_t m2 = (in && v.z >= T1) ? 1u : 0u;
    const uint32_t m3 = (in && v.w >= T1) ? 1u : 0u;
    const uint32_t nm = m0 + m1 + m2 + m3;
    const unsigned long long bb0 = __ballot((int)(nm & 1u));
    const unsigned long long bb1 = __ballot((int)((nm >> 1) & 1u));
    const unsigned long long bb2 = __ballot((int)((nm >> 2) & 1u));
    const uint32_t total = (uint32_t)__popcll(bb0) + 2u * (uint32_t)__popcll(bb1)
                         + 4u * (uint32_t)__popcll(bb2);
    if (total == 0u) return;                   // wave-uniform
    const int lane = tid & 31;
    uint32_t base = 0u;
    if (lane == 0) base = atomicAdd(shcnt, total);
    base = (uint32_t)__shfl((int)base, 0, 32);
    const unsigned long long lt = (1ull << lane) - 1ull;
    uint32_t off = base + (uint32_t)__popcll(bb0 & lt) + 2u * (uint32_t)__popcll(bb1 & lt)
                        + 4u * (uint32_t)__popcll(bb2 & lt);
    const uint32_t e = (uint32_t)(h + (i << 2));
    if (m0) { if (off < CAP) { candKey[off] = v.x; candIdx[off] = e;      } ++off; }
    if (m1) { if (off < CAP) { candKey[off] = v.y; candIdx[off] = e + 1u; } ++off; }
    if (m2) { if (off < CAP) { candKey[off] = v.z; candIdx[off] = e + 2u; } ++off; }
    if (m3) { if (off < CAP) { candKey[off] = v.w; candIdx[off] = e + 3u; } ++off; }
}
__device__ __forceinline__ void collect(const float* data, int h, int n4, int ntail, int t0,
                                        uint32_t T1, uint32_t* shcnt,
                                        uint32_t* candKey, uint32_t* candIdx) {
    const int tid = threadIdx.x;
    if (tid < h) {
        const uint32_t k = __float_as_uint(data[tid]);
        if (k >= T1) { const uint32_t p = atomicAdd(shcnt, 1u);
                       if (p < CAP) { candKey[p] = k; candIdx[p] = (uint32_t)tid; } }
    }
    if (tid < ntail) {
        const uint32_t k = __float_as_uint(data[t0 + tid]);
        if (k >= T1) { const uint32_t p = atomicAdd(shcnt, 1u);
                       if (p < CAP) { candKey[p] = k; candIdx[p] = (uint32_t)(t0 + tid); } }
    }
    const uint4* rb4 = (const uint4*)(data + h);
    for (int k = 0; k < NITER; ++k)
        collect_chunk(k, tid, rb4, h, n4, T1, shcnt, candKey, candIdx);
}

// ---------------- exact rank selection (lowest-index tie break) ----------------
__device__ __forceinline__ void rank_select(uint32_t C2, const uint32_t* kArr,
                                            const uint32_t* iArr, int* topIdx, float* topVal) {
    const int tid = threadIdx.x;
    for (uint32_t j = (uint32_t)tid; j < C2; j += BLOCK) {
        const uint32_t kj = kArr[j], ij = iArr[j];
        uint32_t r = 0;
        for (uint32_t t = 0; t < C2; ++t) {
            const uint32_t kt = kArr[t];
            r += (kt > kj || (kt == kj && iArr[t] < ij)) ? 1u : 0u;
        }
        if (r < (uint32_t)KTOP) { topIdx[r] = (int)ij; topVal[r] = __uint_as_float(kj); }
    }
    __syncthreads();
}

// ---------------- second-level refine (512 bins, candidate space only) + rank ----------------
__device__ __forceinline__ void refine_and_rank(uint32_t C, uint32_t b0, uint32_t above1,
                                                const uint32_t* candKey, const uint32_t* candIdx,
                                                uint32_t* histR,   // 512
                                                uint32_t* grp16,   // 16
                                                uint32_t* c2K, uint32_t* c2I, uint32_t* shw,
                                                int* topIdx, float* topVal) {
    const int tid = threadIdx.x;
    if (C <= CAP2) { rank_select(C, candKey, candIdx, topIdx, topVal); return; }
    if (tid < 512) histR[tid] = 0;
    __syncthreads();
    for (uint32_t i = (uint32_t)tid; i < C; i += BLOCK) {
        const uint32_t k = candKey[i];
        if ((k >> 19) == b0) atomicAdd(&histR[(k >> 10) & 511u], 1u);
    }
    __syncthreads();
    if (tid < 16) {
        uint32_t ssum = 0;
        for (int j = 0; j < 32; ++j) ssum += histR[(tid << 5) + j];
        grp16[tid] = ssum;
    }
    __syncthreads();
    if (tid == 0) {
        const uint32_t need = (uint32_t)KTOP - above1;     // >= 1
        uint32_t acc = 0; int g = 15;
        for (; g > 0; --g) { const uint32_t c = grp16[g]; if (acc + c >= need) break; acc += c; }
        int b = (g << 5) + 31;
        for (; b > (g << 5); --b) { const uint32_t c = histR[b]; if (acc + c >= need) break; acc += c; }
        shw[1] = (b0 << 19) | ((uint32_t)b << 10);         // T2
        shw[0] = 0;
    }
    __syncthreads();
    const uint32_t T2 = shw[1];
    for (uint32_t i = (uint32_t)tid; i < C; i += BLOCK) {
        const uint32_t k = candKey[i];
        if (k >= T2) {
            const uint32_t p = atomicAdd(&shw[0], 1u);
            if (p < CAP2) { c2K[p] = k; c2I[p] = candIdx[i]; }
        }
    }
    __syncthreads();
    uint32_t C2 = shw[0]; if (C2 > CAP2) C2 = CAP2;
    rank_select(C2, c2K, c2I, topIdx, topVal);
}

// ---------------- main kernel: one workgroup per row ----------------
__global__ __launch_bounds__(BLOCK)
void topk_jsd_kernel(const float* __restrict__ P, const float* __restrict__ Q,
                     float* __restrict__ out, int nrows) {
    __shared__ __align__(16) float rowbuf[VV + 4];
    __shared__ uint32_t hist13[8192];
    __shared__ uint32_t grpA[512];
    __shared__ uint32_t grpB[32];
    __shared__ uint32_t histR[512];
    __shared__ uint32_t grp16[16];
    __shared__ uint32_t candKey[CAP];
    __shared__ uint32_t candIdx[CAP];
    __shared__ uint32_t c2K[CAP2];
    __shared__ uint32_t c2I[CAP2];
    __shared__ uint32_t shv[4];
    __shared__ uint32_t shw[2];
    __shared__ int   topPIdx[KTOP]; __shared__ float topPVal[KTOP];
    __shared__ int   topQIdx[KTOP]; __shared__ float topQVal[KTOP];
    __shared__ float qAtP[KTOP], pAtQ[KTOP];
    __shared__ float cp[2 * KTOP], cq[2 * KTOP], terms[2 * KTOP];
    __shared__ float shS[2];

    const int row = blockIdx.x;
    if (row >= nrows) return;
    const int tid = threadIdx.x;
    const float* prow = P + (size_t)row * VV;
    const float* qrow = Q + (size_t)row * VV;
    const uint32_t ldsBase = (uint32_t)(uintptr_t)(&rowbuf[0]);

    // ---- P: issue stream, histogram hidden under the load ----
    const int sP = ((int)((uintptr_t)prow & 15)) >> 2;
    const int hP = (4 - sP) & 3;
    const int n4P = (VV - hP) >> 2;
    const int t0P = hP + (n4P << 2);
    const int ntP = VV - t0P;
    issue_row_loads(prow, ldsBase, sP, hP, n4P, ntP, t0P);
    for (int i = tid; i < 8192; i += BLOCK) hist13[i] = 0;
    __syncthreads();
    {
        const float* data = rowbuf + sP;
        const uint4* rb4 = (const uint4*)(data + hP);
        hist_pipe<0>(tid, data, rb4, hP, n4P, ntP, t0P, hist13);
    }
    __syncthreads();
    scan13(hist13, grpA, grpB, shv);
    collect(rowbuf + sP, hP, n4P, ntP, t0P, shv[2], &shv[3], candKey, candIdx);
    __syncthreads();
    uint32_t CP = shv[3]; if (CP > CAP) CP = CAP;
    const uint32_t b0P = shv[0], aboveP = shv[1];

    // ---- issue Q stream early; finish P while Q flows from HBM ----
    const int sQ = ((int)((uintptr_t)qrow & 15)) >> 2;
    const int hQ = (4 - sQ) & 3;
    const int n4Q = (VV - hQ) >> 2;
    const int t0Q = hQ + (n4Q << 2);
    const int ntQ = VV - t0Q;
    issue_row_loads(qrow, ldsBase, sQ, hQ, n4Q, ntQ, t0Q);

    refine_and_rank(CP, b0P, aboveP, candKey, candIdx, histR, grp16,
                    c2K, c2I, shw, topPIdx, topPVal);
    for (int i = tid; i < 8192; i += BLOCK) hist13[i] = 0;
    __syncthreads();

    // ---- Q: pipelined histogram against its own ASYNCcnt ----
    {
        const float* data = rowbuf + sQ;
        const uint4* rb4 = (const uint4*)(data + hQ);
        hist_pipe<0>(tid, data, rb4, hQ, n4Q, ntQ, t0Q, hist13);
    }
    __syncthreads();
    scan13(hist13, grpA, grpB, shv);
    collect(rowbuf + sQ, hQ, n4Q, ntQ, t0Q, shv[2], &shv[3], candKey, candIdx);
    __syncthreads();
    uint32_t CQ = shv[3]; if (CQ > CAP) CQ = CAP;
    if (tid < KTOP) qAtP[tid] = (rowbuf + sQ)[topPIdx[tid]];   // q at p-top (LDS)
    refine_and_rank(CQ, shv[0], shv[1], candKey, candIdx, histR, grp16,
                    c2K, c2I, shw, topQIdx, topQVal);
    if (tid < KTOP) pAtQ[tid] = prow[topQIdx[tid]];            // p at q-top (global)
    __syncthreads();

    // ---- union contributions (dup q-entries contribute exact 0) ----
    if (tid < 2 * KTOP) {
        if (tid < KTOP) { cp[tid] = topPVal[tid]; cq[tid] = qAtP[tid]; }
        else {
            const int j = tid - KTOP, qi = topQIdx[j];
            bool dup = false;
            for (int i = 0; i < KTOP; ++i) dup |= (topPIdx[i] == qi);
            cp[tid] = dup ? 0.0f : pAtQ[j];
            cq[tid] = dup ? 0.0f : topQVal[j];
        }
    }
    __syncthreads();
    if (tid < 32) {        // wave-parallel deterministic masked sums
        float sp = 0.0f, sq = 0.0f;
        for (int i = tid; i < 2 * KTOP; i += 32) { sp += cp[i]; sq += cq[i]; }
        for (int off = 16; off > 0; off >>= 1) {
            sp += __shfl_down(sp, off, 32);
            sq += __shfl_down(sq, off, 32);
        }
        if (tid == 0) { shS[0] = sp; shS[1] = sq; }
    }
    __syncthreads();
    if (tid < 2 * KTOP) {
        const float pm = cp[tid] / (shS[0] + EPSF);
        const float qm = cq[tid] / (shS[1] + EPSF);
        const float m  = 0.5f * (pm + qm);
        const float ps = pm + EPSF, qs = qm + EPSF, ms = m + EPSF;
        terms[tid] = 0.5f * (ps * logf(ps / ms) + qs * logf(qs / ms));
    }
    __syncthreads();
    if (tid < 32) {        // wave-parallel deterministic JSD sum
        float s = 0.0f;
        for (int i = tid; i < 2 * KTOP; i += 32) s += terms[i];
        for (int off = 16; off > 0; off >>= 1) s += __shfl_down(s, off, 32);
        if (tid == 0) out[row] = s;
    }
}

extern "C" void kernel_launch(void* const* d_in, const int* in_sizes, int n_in,
                              void* d_out, int out_size, void* d_ws, size_t ws_size,
                              hipStream_t stream) {
    (void)in_sizes; (void)n_in; (void)d_ws; (void)ws_size;
    const float* P = (const float*)d_in[0];
    const float* Q = (const float*)d_in[1];
    float* out = (float*)d_out;
    const int nrows = out_size;            // B*S = 4096
    hipLaunchKernelGGL(topk_jsd_kernel, dim3(nrows), dim3(BLOCK), 0, stream,
                       P, Q, out, nrows);
}